// CorrelationLayer_20925080666495
// MI455X (gfx1250) — compile-verified
//
#include <hip/hip_runtime.h>
#include <hip/hip_bf16.h>

#define H_DIM 192
#define W_DIM 256
#define HW (H_DIM*W_DIM)
#define FH 48
#define FW 64
#define FHW (FH*FW)
#define CIN 64
#define COUT 4
#define MH 39
#define MW 51
#define LQ (MH*MW)      /* 1989 */
#define LPAD 2000
#define NPT 125         /* ceil(1989/16), 125*16 == 2000 */
#define NLM 32
#define NWAVES 8
#define NEG_BIG (-3.402823466e38f)

typedef float v2f __attribute__((ext_vector_type(2)));
typedef float v8f __attribute__((ext_vector_type(8)));

__device__ __forceinline__ float sigmoidf(float x){ return 1.0f/(1.0f+__expf(-x)); }
/* STEP arrays: _steps(38) -> [39,37,...,3, 1,3,...,39] (39), _steps(51) -> 51 entries */
__device__ __forceinline__ float step_h_val(int r){ return (float)((r<19)?(39-2*r):(2*(r-19)+1)); }
__device__ __forceinline__ float step_w_val(int c){ return (float)((c<25)?(51-2*c):(2*(c-25)+1)); }

struct Crop { int top, left, dy1, dx1, nh, nw; };
__device__ __forceinline__ Crop crop_coords(int cx, int cy){
  Crop cr;
  int top    = cy - (MH+1)/2;   /* cy-20 */
  int bottom = cy + MH/2;       /* cy+19 */
  int left   = cx - (MW+1)/2;   /* cx-26 */
  int right  = cx + MW/2;       /* cx+25 */
  int dy1 = max(-top,0),  dy2 = max(bottom-H_DIM,0);
  int dx1 = max(-left,0), dx2 = max(right-W_DIM,0);
  cr.top = top+dy1; cr.left = left+dx1;
  cr.nh = max(bottom-dy2-cr.top,  0);
  cr.nw = max(right-dx2-cr.left, 0);
  cr.dy1 = dy1; cr.dx1 = dx1;
  return cr;
}

/* -------- Kernel A: PReLU + (4,64) channel reduce at 48x64 -------- */
__global__ void k_reduce(const float* __restrict__ fea, const float* __restrict__ w,
                         const float* __restrict__ b, const float* __restrict__ a_ptr,
                         float* __restrict__ red){
  int pix = blockIdx.x*blockDim.x + threadIdx.x;
  if (pix >= FHW) return;
  float a = a_ptr[0];
  float s0=0.f,s1=0.f,s2=0.f,s3=0.f;
  for (int c=0;c<CIN;++c){
    float x = fea[c*FHW + pix];
    float pr = fmaxf(x,0.f) + a*fminf(x,0.f);
    s0 += w[0*CIN+c]*pr; s1 += w[1*CIN+c]*pr;
    s2 += w[2*CIN+c]*pr; s3 += w[3*CIN+c]*pr;
  }
  red[0*FHW+pix]=s0+b[0]; red[1*FHW+pix]=s1+b[1];
  red[2*FHW+pix]=s2+b[2]; red[3*FHW+pix]=s3+b[3];
}

/* -------- Kernel B: build K (cloth), P (person), V (loc) patches, [q][c] interleave -------- */
__global__ void k_patch(const float* __restrict__ loc, const float* __restrict__ redc,
                        const float* __restrict__ redp, const float* __restrict__ sp,
                        const int* __restrict__ clm, const int* __restrict__ plm,
                        float* __restrict__ Kc, float* __restrict__ Pp, float* __restrict__ Vv){
  int i = blockIdx.x;
  int ccx = clm[i*2+0], ccy = clm[i*2+1];
  int pcx = plm[i*2+0], pcy = plm[i*2+1];
  bool use_lm = !(((ccx==0)&&(ccy==0)) || ((pcx==0)&&(pcy==0)));
  float adj_cw = sp[i*4+0], adj_ch = sp[i*4+1];
  float adj_pw = sp[i*4+2], adj_ph = sp[i*4+3];
  const float init_h = (float)W_DIM/5.0f;  /* 51.2 */
  const float init_w = (float)H_DIM/5.0f;  /* 38.4 */
  float hC = init_h*sigmoidf(adj_ch), wC = init_w*sigmoidf(adj_cw);
  float hP = init_h*sigmoidf(adj_ph), wP = init_w*sigmoidf(adj_pw);
  Crop cc = crop_coords(ccx, ccy);
  Crop cp = crop_coords(pcx, pcy);
  for (int j = threadIdx.x; j < LPAD; j += blockDim.x){
    size_t kb = (size_t)i*LPAD + j;
    if (j < LQ){
      int r = j / MW, q = j - r*MW;
      { /* cloth + loc (use rows_c/cols_c, valid_c) */
        bool v = use_lm && (r < cc.nh) && (q < cc.nw);
        int row = min(max(cc.top  + r, 0), H_DIM-1);
        int col = min(max(cc.left + q, 0), W_DIM-1);
        int mr  = min(max(cc.dy1 + r, 0), MH-1);
        int mc  = min(max(cc.dx1 + q, 0), MW-1);
        float mval = sigmoidf((hC - step_h_val(mr))*2.f) * sigmoidf((wC - step_w_val(mc))*2.f);
        int fp = (row>>2)*FW + (col>>2);
        #pragma unroll
        for (int c=0;c<COUT;++c)
          Kc[kb*4+c] = v ? redc[c*FHW+fp]*mval : 0.f;
        Vv[kb*2+0] = v ? loc[(size_t)(i*2+0)*HW + row*W_DIM + col] : -1.f;
        Vv[kb*2+1] = v ? loc[(size_t)(i*2+1)*HW + row*W_DIM + col] : -1.f;
      }
      { /* person */
        bool v = use_lm && (r < cp.nh) && (q < cp.nw);
        int row = min(max(cp.top  + r, 0), H_DIM-1);
        int col = min(max(cp.left + q, 0), W_DIM-1);
        int mr  = min(max(cp.dy1 + r, 0), MH-1);
        int mc  = min(max(cp.dx1 + q, 0), MW-1);
        float mval = sigmoidf((hP - step_h_val(mr))*2.f) * sigmoidf((wP - step_w_val(mc))*2.f);
        int fp = (row>>2)*FW + (col>>2);
        #pragma unroll
        for (int c=0;c<COUT;++c)
          Pp[kb*4+c] = v ? redp[c*FHW+fp]*mval : 0.f;
      }
    } else { /* zero the 1989..1999 padding so WMMA pad rows/cols are clean */
      #pragma unroll
      for (int c=0;c<COUT;++c){ Kc[kb*4+c]=0.f; Pp[kb*4+c]=0.f; }
      Vv[kb*2+0]=0.f; Vv[kb*2+1]=0.f;
    }
  }
}

/* -------- Kernel C: WMMA scores + two-pass softmax + flow --------
 * Pass 1: per-lane max over its q subset (no shuffles in loop), one butterfly max.
 * Pass 2: per-lane exp-accumulate (no shuffles in loop), one butterfly sum.
 */
__global__ __launch_bounds__(256) void k_corr(const float* __restrict__ Kc,
                                              const float* __restrict__ Pp,
                                              const float* __restrict__ Vv,
                                              float* __restrict__ Fl){
  __shared__ __align__(16) float Klds[LPAD*4];  /* 32000 B */
  __shared__ __align__(16) float Vlds[LPAD*2];  /* 16000 B */
  const int i = blockIdx.x;
  const int tid = threadIdx.x;
  { /* stage K and V for this landmark into LDS */
    const float4* Ks = (const float4*)(Kc + (size_t)i*LPAD*4);
    float4* Kd = (float4*)Klds;
    for (int t = tid; t < LPAD; t += blockDim.x) Kd[t] = Ks[t];
    const float2* Vs = (const float2*)(Vv + (size_t)i*LPAD*2);
    float2* Vd = (float2*)Vlds;
    for (int t = tid; t < LPAD; t += blockDim.x) Vd[t] = Vs[t];
  }
  __syncthreads();

  const int lane = tid & 31;
  const int wave = tid >> 5;
  const int half = lane >> 4;
  const int n    = lane & 15;
  const int cb   = half*2;   /* k-base for A/B fragments: lanes 0-15 hold k=0,1; 16-31 hold k=2,3 */
  const bool last_ok = (16*(NPT-1) + n) < LQ;   /* pad mask, only relevant in last tile */

  for (int pt = blockIdx.y*NWAVES + wave; pt < NPT; pt += gridDim.y*NWAVES){
    const int p0 = pt*16;
    /* A fragment: A[m][k] = P[k][p0+m], m = lane%16, k = cb..cb+1 (8B load) */
    float2 av = *(const float2*)(Pp + (((size_t)i*LPAD + p0 + n)*4 + cb));
    v2f A; A.x = av.x; A.y = av.y;

    /* ---- Pass 1: per-lane row max (mask-free main loop) ---- */
    float m[8];
    #pragma unroll
    for (int j=0;j<8;++j) m[j] = NEG_BIG;
    for (int qt=0; qt<NPT-1; ++qt){
      const int q = qt*16 + n;
      float2 bv = *(const float2*)&Klds[q*4 + cb];
      v2f B; B.x = bv.x; B.y = bv.y;
      v8f Cacc = {0.f,0.f,0.f,0.f,0.f,0.f,0.f,0.f};
      Cacc = __builtin_amdgcn_wmma_f32_16x16x4_f32(false, A, false, B,
                                                   (short)0, Cacc, false, false);
      #pragma unroll
      for (int j=0;j<8;++j) m[j] = fmaxf(m[j], Cacc[j]);
    }
    { /* last tile: mask pad columns */
      const int q = (NPT-1)*16 + n;
      float2 bv = *(const float2*)&Klds[q*4 + cb];
      v2f B; B.x = bv.x; B.y = bv.y;
      v8f Cacc = {0.f,0.f,0.f,0.f,0.f,0.f,0.f,0.f};
      Cacc = __builtin_amdgcn_wmma_f32_16x16x4_f32(false, A, false, B,
                                                   (short)0, Cacc, false, false);
      #pragma unroll
      for (int j=0;j<8;++j) m[j] = fmaxf(m[j], last_ok ? Cacc[j] : NEG_BIG);
    }
    /* one butterfly max across the 16 lanes of each half (rows j + 8*half) */
    #pragma unroll
    for (int msk=1; msk<16; msk<<=1){
      #pragma unroll
      for (int j=0;j<8;++j) m[j] = fmaxf(m[j], __shfl_xor(m[j], msk, 32));
    }

    /* ---- Pass 2: per-lane exp accumulate (mask-free main loop) ---- */
    float s[8], f0[8], f1[8];
    #pragma unroll
    for (int j=0;j<8;++j){ s[j]=0.f; f0[j]=0.f; f1[j]=0.f; }
    for (int qt=0; qt<NPT-1; ++qt){
      const int q = qt*16 + n;
      float2 bv = *(const float2*)&Klds[q*4 + cb];
      v2f B; B.x = bv.x; B.y = bv.y;
      v8f Cacc = {0.f,0.f,0.f,0.f,0.f,0.f,0.f,0.f};
      Cacc = __builtin_amdgcn_wmma_f32_16x16x4_f32(false, A, false, B,
                                                   (short)0, Cacc, false, false);
      float2 vv = *(const float2*)&Vlds[q*2];
      #pragma unroll
      for (int j=0;j<8;++j){
        float e = __expf(Cacc[j] - m[j]);
        s[j]  += e;
        f0[j] += e*vv.x;
        f1[j] += e*vv.y;
      }
    }
    { /* last tile: e = 0 on pad columns */
      const int q = (NPT-1)*16 + n;
      float2 bv = *(const float2*)&Klds[q*4 + cb];
      v2f B; B.x = bv.x; B.y = bv.y;
      v8f Cacc = {0.f,0.f,0.f,0.f,0.f,0.f,0.f,0.f};
      Cacc = __builtin_amdgcn_wmma_f32_16x16x4_f32(false, A, false, B,
                                                   (short)0, Cacc, false, false);
      float2 vv = *(const float2*)&Vlds[q*2];
      #pragma unroll
      for (int j=0;j<8;++j){
        float e = last_ok ? __expf(Cacc[j] - m[j]) : 0.f;
        s[j]  += e;
        f0[j] += e*vv.x;
        f1[j] += e*vv.y;
      }
    }
    /* one butterfly sum across the 16 lanes of each half */
    #pragma unroll
    for (int msk=1; msk<16; msk<<=1){
      #pragma unroll
      for (int j=0;j<8;++j){
        s[j]  += __shfl_xor(s[j],  msk, 32);
        f0[j] += __shfl_xor(f0[j], msk, 32);
        f1[j] += __shfl_xor(f1[j], msk, 32);
      }
    }
    if (n == 0){
      #pragma unroll
      for (int j=0;j<8;++j){
        int p = p0 + half*8 + j;
        if (p < LQ){
          float inv = 1.f / s[j];
          Fl[((size_t)i*LPAD + p)*2 + 0] = f0[j]*inv;
          Fl[((size_t)i*LPAD + p)*2 + 1] = f1[j]*inv;
        }
      }
    }
  }
}

/* -------- Kernel D: fill outputs (flow=-1, mask=0) -------- */
__global__ void k_fill(float* __restrict__ out, int total){
  int idx = blockIdx.x*blockDim.x + threadIdx.x;
  if (idx >= total) return;
  out[idx] = (idx < 64*HW) ? -1.0f : 0.0f;
}

/* -------- Kernel E: scatter flow + mask at person-patch positions -------- */
__global__ void k_scatter(const float* __restrict__ Fl, const int* __restrict__ clm,
                          const int* __restrict__ plm, float* __restrict__ out){
  int i = blockIdx.x;
  int ccx = clm[i*2+0], ccy = clm[i*2+1];
  int pcx = plm[i*2+0], pcy = plm[i*2+1];
  bool use_lm = !(((ccx==0)&&(ccy==0)) || ((pcx==0)&&(pcy==0)));
  if (!use_lm) return;
  Crop cp = crop_coords(pcx, pcy);
  float* flow = out;
  float* msk  = out + (size_t)64*HW;
  for (int j = threadIdx.x; j < LQ; j += blockDim.x){
    int r = j / MW, q = j - r*MW;
    if (r < cp.nh && q < cp.nw){
      int row = cp.top  + r;   /* in-range by construction for valid entries */
      int col = cp.left + q;
      size_t o = (size_t)row*W_DIM + col;
      flow[(size_t)(i*2+0)*HW + o] = Fl[((size_t)i*LPAD + j)*2 + 0];
      flow[(size_t)(i*2+1)*HW + o] = Fl[((size_t)i*LPAD + j)*2 + 1];
      msk[(size_t)i*HW + o] = 1.0f;
    }
  }
}

extern "C" void kernel_launch(void* const* d_in, const int* in_sizes, int n_in,
                              void* d_out, int out_size, void* d_ws, size_t ws_size,
                              hipStream_t stream){
  (void)in_sizes; (void)n_in; (void)out_size; (void)ws_size;
  const float* loc   = (const float*)d_in[0];
  const float* fea_c = (const float*)d_in[1];
  const float* fea_p = (const float*)d_in[2];
  const float* sp    = (const float*)d_in[3];
  /* d_in[4]=H, d_in[5]=W: fixed 192x256, hardcoded */
  const int*   clm   = (const int*)d_in[6];
  const int*   plm   = (const int*)d_in[7];
  const float* w1    = (const float*)d_in[8];
  const float* b1    = (const float*)d_in[9];
  const float* a1    = (const float*)d_in[10];
  const float* w2    = (const float*)d_in[11];
  const float* b2    = (const float*)d_in[12];
  const float* a2    = (const float*)d_in[13];

  float* ws   = (float*)d_ws;
  float* redc = ws;                               /* 4*3072            */
  float* redp = redc + 4*FHW;                     /* 4*3072            */
  float* Kc   = redp + 4*FHW;                     /* 32*2000*4         */
  float* Pp   = Kc   + (size_t)NLM*LPAD*4;        /* 32*2000*4         */
  float* Vv   = Pp   + (size_t)NLM*LPAD*4;        /* 32*2000*2         */
  float* Fl   = Vv   + (size_t)NLM*LPAD*2;        /* 32*2000*2         */
  float* out  = (float*)d_out;

  k_reduce<<<dim3((FHW+255)/256), 256, 0, stream>>>(fea_c, w1, b1, a1, redc);
  k_reduce<<<dim3((FHW+255)/256), 256, 0, stream>>>(fea_p, w2, b2, a2, redp);
  k_patch <<<dim3(NLM),           256, 0, stream>>>(loc, redc, redp, sp, clm, plm, Kc, Pp, Vv);
  k_corr  <<<dim3(NLM, 8),        256, 0, stream>>>(Kc, Pp, Vv, Fl);
  const int total = 96*HW;
  k_fill  <<<dim3((total+255)/256), 256, 0, stream>>>(out, total);
  k_scatter<<<dim3(NLM),          256, 0, stream>>>(Fl, clm, plm, out);
}